// SegmentedPolynomialFromUniform1dJit_1151051235364
// MI455X (gfx1250) — compile-verified
//
#include <hip/hip_runtime.h>

// CDNA5 (gfx1250) segmented-polynomial kernel.
// out[k,c] per edge = Cmat(16x256) x XY(256x64) GEMM via V_WMMA_F32_16X16X4_F32.

typedef float v2f __attribute__((ext_vector_type(2)));
typedef float v8f __attribute__((ext_vector_type(8)));

#define SEG 16
#define EXTC 64
#define ROW (SEG * EXTC)     // 1024 floats per node/edge row
#define NKCH 64              // K-chunks of 4 over ij=256
#define EPB 8                // edges per workgroup

// ---- Kernel 1: densify paths into Cg[k*256 + i*16 + j] (fp32, accumulate dups)
__global__ void build_cmat_kernel(const int* __restrict__ pidx,
                                  const float* __restrict__ pcoef,
                                  float* __restrict__ Cg, int P) {
    int p = blockIdx.x * blockDim.x + threadIdx.x;
    if (p < P) {
        int i = pidx[3 * p + 0];
        int j = pidx[3 * p + 1];
        int k = pidx[3 * p + 2];
        atomicAdd(&Cg[k * 256 + i * 16 + j], pcoef[p]);
    }
}

// ---- Kernel 2: per-edge GEMM with fp32 WMMA + scatter-add
__global__ __launch_bounds__(128)
void seg_poly_wmma_kernel(const float* __restrict__ x,
                          const float* __restrict__ y,
                          const int* __restrict__ idx_in,
                          const int* __restrict__ idx_out,
                          const float* __restrict__ Cg,
                          float* __restrict__ out,
                          int Etot) {
    // LDS: Cmat pre-swizzled into A-fragment order.
    // Chunk r covers ij = 4r..4r+3 (i = r>>2 constant per chunk, j = (r&3)*4 + 0..3).
    // A 16x4 fp32 fragment layout: lane L holds row m=L%16; half h=L/16 selects
    // K-pair {2h, 2h+1}. LDS float word w = r*64 + L*2 + d  <-  Cg[m*256 + 4r + 2h + d].
    __shared__ float sC[NKCH * 32 * 2];   // 16 KB

    const int tid  = threadIdx.x;
    const int lane = tid & 31;
    const int wave = tid >> 5;            // 4 waves: 16-channel column tiles
    const int c    = lane & 15;           // channel within tile
    const int h    = lane >> 4;           // half-wave (K-pair / M-half select)
    const int ch0  = wave * 16;

    for (int w = tid; w < NKCH * 64; w += 128) {
        int r  = w >> 6;
        int rm = w & 63;
        int L  = rm >> 1;
        int d  = rm & 1;
        int m  = L & 15;
        int hh = L >> 4;
        sC[w] = Cg[m * 256 + 4 * r + 2 * hh + d];
    }
    __syncthreads();

    const int e0 = blockIdx.x * EPB;

    for (int t = 0; t < EPB; ++t) {
        const int e = e0 + t;
        if (e >= Etot) break;            // workgroup-uniform

        const int nin  = idx_in[e];
        const int nout = idx_out[e];

        // Per-lane tiles: x[seg, ch0+c] (replicated across half-waves), same for y.
        const float* __restrict__ xrow = x + (size_t)nin * ROW + ch0 + c;
        const float* __restrict__ yrow = y + (size_t)e   * ROW + ch0 + c;
        float xv[SEG], yv[SEG];
#pragma unroll
        for (int s = 0; s < SEG; ++s) {
            xv[s] = xrow[s * EXTC];
            yv[s] = yrow[s * EXTC];
        }

        v8f acc = {};
#pragma unroll
        for (int r = 0; r < NKCH; ++r) {
            // A fragment: ds_load_b64 from pre-swizzled LDS
            const v2f a = *reinterpret_cast<const v2f*>(&sC[(r * 32 + lane) * 2]);
            // B fragment built on the fly: XY[4r + 2h + {0,1}, ch]
            const int i  = r >> 2;          // constant per unrolled chunk
            const int j0 = (r & 3) * 4;     // constant per unrolled chunk
            v2f b;
            b.x = xv[i] * (h ? yv[j0 + 2] : yv[j0 + 0]);
            b.y = xv[i] * (h ? yv[j0 + 3] : yv[j0 + 1]);
            // D = A(16x4) * B(4x16) + C, fp32 XDL path
            acc = __builtin_amdgcn_wmma_f32_16x16x4_f32(
                /*neg_a=*/false, a, /*neg_b=*/false, b,
                /*c_mod=*/(short)0, acc, /*reuse_a=*/false, /*reuse_b=*/false);
        }

        // Scatter-add the 16x16 tile: lane holds out[k = v + 8h][ch0 + c].
        float* __restrict__ orow = out + (size_t)nout * ROW + ch0 + c;
#pragma unroll
        for (int v = 0; v < 8; ++v) {
            atomicAdd(&orow[(v + 8 * h) * EXTC], acc[v]);   // global_atomic_add_f32 (L2-resident)
        }
    }
}

extern "C" void kernel_launch(void* const* d_in, const int* in_sizes, int n_in,
                              void* d_out, int out_size, void* d_ws, size_t ws_size,
                              hipStream_t stream) {
    const float* x       = (const float*)d_in[0];
    const float* y       = (const float*)d_in[1];
    const int*   idx_in  = (const int*)d_in[2];
    const int*   idx_out = (const int*)d_in[3];
    const int*   pidx    = (const int*)d_in[4];
    const float* pcoef   = (const float*)d_in[5];
    float*       out     = (float*)d_out;
    float*       Cg      = (float*)d_ws;    // 4096 floats = 16 KB

    const int E = in_sizes[2];
    const int P = in_sizes[5];

    // Zero scatter target and dense coefficient tensor (graph-capturable).
    hipMemsetAsync(d_out, 0, (size_t)out_size * sizeof(float), stream);
    hipMemsetAsync(d_ws, 0, 4096 * sizeof(float), stream);

    build_cmat_kernel<<<(P + 255) / 256, 256, 0, stream>>>(pidx, pcoef, Cg, P);

    const int grid = (E + EPB - 1) / EPB;   // 2048 workgroups x 4 waves
    seg_poly_wmma_kernel<<<grid, 128, 0, stream>>>(x, y, idx_in, idx_out, Cg, out, E);
}